// GRUDFeatureProcessor_67877663146572
// MI455X (gfx1250) — compile-verified
//
#include <hip/hip_runtime.h>
#include <hip/hip_bf16.h>
#include <stdint.h>

#define B_   64
#define T_   2048
#define D_   64
#define BT_  (B_ * T_)
#define ROWF 256          // 4*D floats per output row

typedef __attribute__((ext_vector_type(2))) float v2f;
typedef __attribute__((ext_vector_type(8))) float v8f;

// ---------------------------------------------------------------------------
// Kernel 1: parallel linear-recurrence scan (unchanged from round 1).
//   beta_t = a_t*beta_{t-1} + c_t,  (a,c) o (a',c') = (a*a', a*c' + c)
// One 256-thread block per (f,b) chain.
// ---------------------------------------------------------------------------
__global__ __launch_bounds__(256) void grud_beta_scan(
    const int* __restrict__ cat1, const int* __restrict__ cat2,
    const float* __restrict__ num1, const float* __restrict__ num2,
    const float* __restrict__ times, float* __restrict__ beta)
{
    const int chain = blockIdx.x;          // 0..255 = f*64 + b
    const int f = chain >> 6;
    const int b = chain & 63;
    const int tid = threadIdx.x;
    const int t0 = tid * 8;
    const float* tm = times + (size_t)b * T_;

    float la[8], lc[8];
    float A = 1.f, C = 0.f;                // identity transform
    #pragma unroll
    for (int k = 0; k < 8; ++k) {
        const int t = t0 + k;
        float a_k, c_k;
        if (t == 0) {
            a_k = 0.f; c_k = 0.f;          // forces beta_0 = 0 (init state)
        } else {
            const float ti = tm[t];
            const float tp = tm[t - 1];
            float mv;
            if (f == 0)      { int   v = cat1[(size_t)b * T_ + t - 1]; mv = (v != -1  && v != 0 ) ? 1.f : 0.f; }
            else if (f == 1) { int   v = cat2[(size_t)b * T_ + t - 1]; mv = (v != -1  && v != 0 ) ? 1.f : 0.f; }
            else if (f == 2) { float v = num1[(size_t)b * T_ + t - 1]; mv = (v != -1.f && v != 0.f) ? 1.f : 0.f; }
            else             { float v = num2[(size_t)b * T_ + t - 1]; mv = (v != -1.f && v != 0.f) ? 1.f : 0.f; }
            const float p = (ti != -1.f) ? 1.f : 0.f;
            a_k = p * mv;
            c_k = p * (ti - tp);
        }
        A = a_k * A;
        C = a_k * C + c_k;
        la[k] = A; lc[k] = C;              // inclusive transform within chunk
    }

    __shared__ float sA[256];
    __shared__ float sC[256];
    sA[tid] = A; sC[tid] = C;
    __syncthreads();
    #pragma unroll
    for (int off = 1; off < 256; off <<= 1) {
        float pa = 1.f, pc = 0.f;
        if (tid >= off) { pa = sA[tid - off]; pc = sC[tid - off]; }
        const float ca = sA[tid], cc = sC[tid];
        __syncthreads();
        sA[tid] = ca * pa;                 // compose(prev, cur)
        sC[tid] = ca * pc + cc;
        __syncthreads();
    }
    const float betaIn = (tid == 0) ? 0.f : sC[tid - 1];

    float* out = beta + (size_t)chain * T_ + t0;
    #pragma unroll
    for (int k = 0; k < 8; ++k)
        out[k] = la[k] * betaIn + lc[k];
}

// ---------------------------------------------------------------------------
// Kernel 2: row writer, one wave per 16 consecutive output rows.
//  * categorical embeddings: global -> LDS -> global via CDNA5 async mover
//  * numeric embeddings: nemb = [num,1] @ [w;b]  done on the matrix pipe with
//    V_WMMA_F32_16X16X4_F32 (A: 16x4 = [num | 1 | 0 | 0], B: 4x16 = [w;b;0;0])
//  * delta/mask: broadcast float4 stores
// ---------------------------------------------------------------------------
__global__ __launch_bounds__(256) void grud_writer(
    const int* __restrict__ cat1, const int* __restrict__ cat2,
    const float* __restrict__ num1, const float* __restrict__ num2,
    const float* __restrict__ times,
    const float* __restrict__ E1, const float* __restrict__ E2,
    const float* __restrict__ w1, const float* __restrict__ b1,
    const float* __restrict__ w2, const float* __restrict__ b2,
    const float* __restrict__ beta,
    float* __restrict__ outF, float* __restrict__ outT,
    float* __restrict__ outD, float* __restrict__ outM)
{
    const int lane      = threadIdx.x & 31;
    const int waveInBlk = threadIdx.x >> 5;
    const int wavesBlk  = blockDim.x >> 5;
    const int waveId    = blockIdx.x * wavesBlk + waveInBlk;
    const int numWaves  = gridDim.x * wavesBlk;

    __shared__ float sbuf[8 * 16 * 128];            // 8 waves * 16 rows * 512B
    const unsigned ldsWave =
        (unsigned)(uintptr_t)(&sbuf[waveInBlk * 16 * 128]);

    const bool loLane = (lane < 16);
    const int  j      = lane & 15;

    // B operands for the nemb WMMAs: row K=0 = w, K=1 = bias, K=2,3 = 0
    // (4x16 f32 B layout: lanes 0-15 hold K=0..1 in vgpr0..1, lanes 16-31 K=2..3)
    v2f Bw1[4], Bw2[4];
    #pragma unroll
    for (int c = 0; c < 4; ++c) {
        Bw1[c].x = loLane ? w1[c * 16 + j] : 0.f;
        Bw1[c].y = loLane ? b1[c * 16 + j] : 0.f;
        Bw2[c].x = loLane ? w2[c * 16 + j] : 0.f;
        Bw2[c].y = loLane ? b2[c * 16 + j] : 0.f;
    }

    const int mOff = loLane ? 0 : 8;                // D-tile row offset per lane half
    const int numGroups = BT_ / 16;                 // 8192 groups of 16 rows

    for (int grp = waveId; grp < numGroups; grp += numWaves) {
        const int rowBase = grp * 16;

        // previous group's async stores must be done reading our LDS slab
        asm volatile("s_wait_asynccnt 0x0" ::: "memory");

        // ---- phase 1: per-row scalars + async embedding loads + delta/mask
        #pragma unroll 4
        for (int r = 0; r < 16; ++r) {
            const int   row = rowBase + r;
            const int   c1  = cat1[row];
            const int   c2  = cat2[row];
            const float n1  = num1[row];
            const float n2  = num2[row];

            // lanes 0-15: E1 row, lanes 16-31: E2 row -> LDS (512B per row)
            {
                const float* srcp = loLane
                    ? (E1 + ((size_t)c1 << 6) + ((size_t)j << 2))
                    : (E2 + ((size_t)c2 << 6) + ((size_t)j << 2));
                unsigned long long gsrc = (unsigned long long)(uintptr_t)srcp;
                unsigned ldsOff = ldsWave + (unsigned)(r * 512 + lane * 16);
                asm volatile("global_load_async_to_lds_b128 %0, %1, off"
                             :: "v"(ldsOff), "v"(gsrc) : "memory");
            }

            // delta: broadcast beta[f,b,t] across D
            {
                const float bA = beta[(size_t)(lane >> 4) * BT_ + row];        // f=0/1
                const float bB = beta[(size_t)((lane >> 4) + 2) * BT_ + row];  // f=2/3
                float4* dRow = (float4*)(outD + (size_t)row * ROWF);
                dRow[lane]      = make_float4(bA, bA, bA, bA);
                dRow[lane + 32] = make_float4(bB, bB, bB, bB);
            }

            // mask: predicates broadcast across D (float 0/1)
            {
                const float m0 = (c1 != -1  && c1 != 0 ) ? 1.f : 0.f;
                const float m1 = (c2 != -1  && c2 != 0 ) ? 1.f : 0.f;
                const float m2 = (n1 != -1.f && n1 != 0.f) ? 1.f : 0.f;
                const float m3 = (n2 != -1.f && n2 != 0.f) ? 1.f : 0.f;
                const float mA = loLane ? m0 : m1;
                const float mB = loLane ? m2 : m3;
                float4* mRow = (float4*)(outM + (size_t)row * ROWF);
                mRow[lane]      = make_float4(mA, mA, mA, mA);
                mRow[lane + 32] = make_float4(mB, mB, mB, mB);
            }

            if (lane == 0) outT[row] = times[row];
        }

        // ---- phase 2: numeric embeddings on the matrix pipe
        // A (16x4 f32): lanes 0-15 carry M=lane, K=0 -> num, K=1 -> 1.0;
        // lanes 16-31 carry K=2,3 -> 0.
        {
            const float an1 = num1[rowBase + j];
            const float an2 = num2[rowBase + j];
            v2f A1, A2;
            A1.x = loLane ? an1 : 0.f;  A1.y = loLane ? 1.f : 0.f;
            A2.x = loLane ? an2 : 0.f;  A2.y = loLane ? 1.f : 0.f;

            float* nembBase = outF + (size_t)rowBase * ROWF + 128
                              + (size_t)mOff * ROWF + j;     // per-lane (M0,N)
            #pragma unroll
            for (int c = 0; c < 4; ++c) {
                v8f acc = {};
                v8f d1 = __builtin_amdgcn_wmma_f32_16x16x4_f32(
                    false, A1, false, Bw1[c], (short)0, acc, false, false);
                v8f d2 = __builtin_amdgcn_wmma_f32_16x16x4_f32(
                    false, A2, false, Bw2[c], (short)0, acc, false, false);
                float* p1 = nembBase + c * 16;                // nemb1 cols
                float* p2 = nembBase + 64 + c * 16;           // nemb2 cols
                #pragma unroll
                for (int r2 = 0; r2 < 8; ++r2) {
                    p1[(size_t)r2 * ROWF] = d1[r2];           // row M = r2 + mOff
                    p2[(size_t)r2 * ROWF] = d2[r2];
                }
            }
        }

        // ---- phase 3: gathered embedding columns LDS -> HBM
        asm volatile("s_wait_asynccnt 0x0" ::: "memory");
        #pragma unroll 4
        for (int r = 0; r < 16; ++r) {
            float* fRow = outF + (size_t)(rowBase + r) * ROWF;
            unsigned long long gdst =
                (unsigned long long)(uintptr_t)(fRow + ((size_t)lane << 2));
            unsigned ldsOff = ldsWave + (unsigned)(r * 512 + lane * 16);
            asm volatile("global_store_async_from_lds_b128 %0, %1, off"
                         :: "v"(gdst), "v"(ldsOff) : "memory");
        }
    }
    asm volatile("s_wait_asynccnt 0x0" ::: "memory");
}

// ---------------------------------------------------------------------------
extern "C" void kernel_launch(void* const* d_in, const int* in_sizes, int n_in,
                              void* d_out, int out_size, void* d_ws, size_t ws_size,
                              hipStream_t stream)
{
    (void)in_sizes; (void)n_in; (void)out_size; (void)ws_size;

    const int*   cat1  = (const int*)  d_in[0];
    const int*   cat2  = (const int*)  d_in[1];
    const float* num1  = (const float*)d_in[2];
    const float* num2  = (const float*)d_in[3];
    const float* times = (const float*)d_in[4];
    const float* E1    = (const float*)d_in[5];
    const float* E2    = (const float*)d_in[6];
    const float* w1    = (const float*)d_in[7];
    const float* b1    = (const float*)d_in[8];
    const float* w2    = (const float*)d_in[9];
    const float* b2    = (const float*)d_in[10];

    float* beta = (float*)d_ws;                       // [4][B][T] = 2 MB

    float* outF = (float*)d_out;                      // [B,T,256]
    float* outT = outF + (size_t)BT_ * ROWF;          // [B,T]
    float* outD = outT + BT_;                         // [B,T,256]
    float* outM = outD + (size_t)BT_ * ROWF;          // [B,T,256]

    grud_beta_scan<<<4 * B_, 256, 0, stream>>>(cat1, cat2, num1, num2, times, beta);

    grud_writer<<<1024, 256, 0, stream>>>(cat1, cat2, num1, num2, times,
                                          E1, E2, w1, b1, w2, b2, beta,
                                          outF, outT, outD, outM);
}